// Model_76081050682085
// MI455X (gfx1250) — compile-verified
//
#include <hip/hip_runtime.h>
#include <hip/hip_bf16.h>
#include <math.h>

#define HD 32

typedef __attribute__((ext_vector_type(2))) float v2f;
typedef __attribute__((ext_vector_type(4))) float v4f;
typedef __attribute__((ext_vector_type(8))) float v8f;

// ---------------------------------------------------------------------------
// float atomic max via sign-aware integer atomics (works with -inf seed)
// ---------------------------------------------------------------------------
__device__ __forceinline__ void atomicMaxFloat(float* addr, float val) {
    if (val >= 0.0f) {
        atomicMax((int*)addr, __float_as_int(val));
    } else {
        atomicMin((unsigned int*)addr, (unsigned int)__float_as_int(val));
    }
}

// ---------------------------------------------------------------------------
// Fill p with -1e9 using non-temporal B128 stores (400 MB stream, keep out of L2)
// ---------------------------------------------------------------------------
__global__ void fill_p_kernel(float* __restrict__ p, long long nElem) {
    long long n4 = nElem >> 2;
    v4f v = {-1e9f, -1e9f, -1e9f, -1e9f};
    v4f* p4 = (v4f*)p;
    long long stride = (long long)gridDim.x * blockDim.x;
    for (long long i = (long long)blockIdx.x * blockDim.x + threadIdx.x; i < n4; i += stride)
        __builtin_nontemporal_store(v, p4 + i);
    // tail (nElem may not be multiple of 4 in general)
    long long tail = n4 << 2;
    for (long long i = tail + (long long)blockIdx.x * blockDim.x + threadIdx.x; i < nElem; i += stride)
        p[i] = -1e9f;
}

// ---------------------------------------------------------------------------
// Init scatter-max buffer to -inf and zero the h_new column-sum accumulator
// ---------------------------------------------------------------------------
__global__ void init_m_kernel(float* __restrict__ mbuf, float* __restrict__ hsum, int nElem) {
    int i = blockIdx.x * blockDim.x + threadIdx.x;
    if (i < nElem) mbuf[i] = -INFINITY;
    if (i < HD) hsum[i] = 0.0f;
}

// ---------------------------------------------------------------------------
// Encoder: z = cat[x, h] @ W_enc + b_enc     (N x 33) @ (33 x 32)
// ---------------------------------------------------------------------------
__global__ void enc_kernel(const float* __restrict__ x, const float* __restrict__ h,
                           const float* __restrict__ We, const float* __restrict__ be,
                           float* __restrict__ z, int n) {
    __shared__ float sW[33 * HD];
    __shared__ float sb[HD];
    for (int i = threadIdx.x; i < 33 * HD; i += blockDim.x) sW[i] = We[i];
    if (threadIdx.x < HD) sb[threadIdx.x] = be[threadIdx.x];
    __syncthreads();
    int i = blockIdx.x * blockDim.x + threadIdx.x;
    if (i >= n) return;
    float xv = x[i];
    float hv[HD];
#pragma unroll
    for (int k = 0; k < HD; ++k) hv[k] = h[(size_t)i * HD + k];
#pragma unroll 4
    for (int c = 0; c < HD; ++c) {
        float acc = sb[c] + xv * sW[c];
#pragma unroll
        for (int k = 0; k < HD; ++k) acc = fmaf(hv[k], sW[(1 + k) * HD + c], acc);
        z[(size_t)i * HD + c] = acc;
    }
}

// ---------------------------------------------------------------------------
// Message GEMM on the matrix pipe: per wave, 16 edges x 32 outputs.
//   messages = z[dest] @ Wm[0:32] + z[src] @ Wm[32:64] + w*Wm[64] + bm
// computed with V_WMMA_F32_16X16X4_F32 (full fp32), then scatter-max into mbuf.
//
// Fragment layout (CDNA5 ISA 7.12.2):
//   A 16x4 f32 : lanes 0-15 hold {K=0,K=1}, lanes 16-31 hold {K=2,K=3}, M = lane&15
//   B 4x16 f32 : row-striped; vreg j, half hh -> K = j + 2*hh, column N = lane&15
//   C/D 16x16  : vreg i, half hh -> M = i + 8*hh, column N = lane&15
// ---------------------------------------------------------------------------
__global__ void msg_kernel(const int* __restrict__ srcs, const int* __restrict__ dsts,
                           const float* __restrict__ ew, const float* __restrict__ z,
                           const float* __restrict__ Wm, const float* __restrict__ bm,
                           float* __restrict__ mbuf, int E_) {
    __shared__ float sW[65 * HD];
    __shared__ float sb[HD];
    for (int i = threadIdx.x; i < 65 * HD; i += blockDim.x) sW[i] = Wm[i];
    if (threadIdx.x < HD) sb[threadIdx.x] = bm[threadIdx.x];
    __syncthreads();

    const int lane  = threadIdx.x & 31;
    const int wave  = threadIdx.x >> 5;
    const int tile  = blockIdx.x * (blockDim.x >> 5) + wave;
    const int ebase = tile * 16;
    if (ebase >= E_) return;

    const int half = lane >> 4;   // 0: K even pair / rows 0-7;  1: K odd pair / rows 8-15
    const int lm   = lane & 15;
    const int n0   = lm;          // column in tile 0
    const int n1   = 16 + lm;     // column in tile 1

    int eSelf = ebase + lm;       // this lane supplies A rows for edge eSelf
    if (eSelf >= E_) eSelf = E_ - 1;
    const int nd = dsts[eSelf];
    const int ns = srcs[eSelf];

    // Seed accumulators with bias + edge-weight rank-1 term.
    v8f c0, c1;
    const float b0v = sb[n0], b1v = sb[n1];
    const float w64n0 = sW[64 * HD + n0], w64n1 = sW[64 * HD + n1];
#pragma unroll
    for (int i = 0; i < 8; ++i) {
        int ee = ebase + i + half * 8;
        float we = ew[ee < E_ ? ee : (E_ - 1)];
        c0[i] = fmaf(we, w64n0, b0v);
        c1[i] = fmaf(we, w64n1, b1v);
    }

    // Two K=32 passes: dest rows vs Wm[0:32], src rows vs Wm[32:64].
#pragma unroll
    for (int pass = 0; pass < 2; ++pass) {
        const float* zrow = z + (size_t)(pass ? ns : nd) * HD;
        const int wbase = pass * HD;
#pragma unroll
        for (int kk = 0; kk < 8; ++kk) {
            const int k0 = kk * 4;
            const int kb = k0 + half * 2;
            v2f a, bf0, bf1;
            a[0]   = zrow[kb + 0];
            a[1]   = zrow[kb + 1];
            bf0[0] = sW[(wbase + kb + 0) * HD + n0];
            bf0[1] = sW[(wbase + kb + 1) * HD + n0];
            bf1[0] = sW[(wbase + kb + 0) * HD + n1];
            bf1[1] = sW[(wbase + kb + 1) * HD + n1];
            c0 = __builtin_amdgcn_wmma_f32_16x16x4_f32(false, a, false, bf0, (short)0, c0, false, false);
            c1 = __builtin_amdgcn_wmma_f32_16x16x4_f32(false, a, false, bf1, (short)0, c1, false, false);
        }
    }

    // Scatter-max into destination nodes.
#pragma unroll
    for (int i = 0; i < 8; ++i) {
        int ee = ebase + i + half * 8;
        if (ee < E_) {
            int de = dsts[ee];
            atomicMaxFloat(&mbuf[(size_t)de * HD + n0], c0[i]);
            atomicMaxFloat(&mbuf[(size_t)de * HD + n1], c1[i]);
        }
    }
}

// ---------------------------------------------------------------------------
// Update + decode + mean partials:
//   h_new = cat[z, m(-inf->0)] @ W_upd + b_upd
//   y     = cat[z, h_new] @ W_dec + b_dec
//   hsum += column sums of h_new (block-reduced)
// ---------------------------------------------------------------------------
__global__ void upd_kernel(const float* __restrict__ z, const float* __restrict__ mbuf,
                           const float* __restrict__ Wu, const float* __restrict__ bu,
                           const float* __restrict__ Wd, const float* __restrict__ bd,
                           float* __restrict__ hnew, float* __restrict__ y,
                           float* __restrict__ hsum, int n) {
    __shared__ float sW[64 * HD];
    __shared__ float sb[HD];
    __shared__ float sWd[64];
    __shared__ float ssum[HD];
    for (int i = threadIdx.x; i < 64 * HD; i += blockDim.x) sW[i] = Wu[i];
    if (threadIdx.x < HD) sb[threadIdx.x] = bu[threadIdx.x];
    if (threadIdx.x < 64) sWd[threadIdx.x] = Wd[threadIdx.x];
    if (threadIdx.x < HD) ssum[threadIdx.x] = 0.0f;
    __syncthreads();

    int i = blockIdx.x * blockDim.x + threadIdx.x;
    if (i < n) {
        float zr[HD], mr[HD];
#pragma unroll
        for (int k = 0; k < HD; ++k) {
            zr[k] = z[(size_t)i * HD + k];
            float mv = mbuf[(size_t)i * HD + k];
            mr[k] = (mv == -INFINITY) ? 0.0f : mv;   // isneginf -> 0
        }
        float yacc = bd[0];
#pragma unroll 2
        for (int c = 0; c < HD; ++c) {
            float acc = sb[c];
#pragma unroll
            for (int k = 0; k < HD; ++k) acc = fmaf(zr[k], sW[k * HD + c], acc);
#pragma unroll
            for (int k = 0; k < HD; ++k) acc = fmaf(mr[k], sW[(HD + k) * HD + c], acc);
            hnew[(size_t)i * HD + c] = acc;
            yacc = fmaf(zr[c], sWd[c], yacc);
            yacc = fmaf(acc, sWd[HD + c], yacc);
            atomicAdd(&ssum[c], acc);
        }
        y[i] = yacc;
    }
    __syncthreads();
    if (threadIdx.x < HD) atomicAdd(&hsum[threadIdx.x], ssum[threadIdx.x]);
}

// ---------------------------------------------------------------------------
// Predecessor edge scores scattered into dense p.
// ---------------------------------------------------------------------------
__global__ void pred_kernel(const int* __restrict__ srcs, const int* __restrict__ dsts,
                            const float* __restrict__ ew, const float* __restrict__ hnew,
                            const float* __restrict__ Wp, const float* __restrict__ bp,
                            float* __restrict__ p, int E_, int n) {
    __shared__ float sW[65];
    __shared__ float sbp;
    if (threadIdx.x < 65) sW[threadIdx.x] = Wp[threadIdx.x];
    if (threadIdx.x == 0) sbp = bp[0];
    __syncthreads();
    int e = blockIdx.x * blockDim.x + threadIdx.x;
    if (e >= E_) return;
    int d = dsts[e], s = srcs[e];
    float acc = sbp + ew[e] * sW[64];
#pragma unroll
    for (int k = 0; k < HD; ++k) acc = fmaf(hnew[(size_t)d * HD + k], sW[k], acc);
#pragma unroll
    for (int k = 0; k < HD; ++k) acc = fmaf(hnew[(size_t)s * HD + k], sW[HD + k], acc);
    p[(size_t)d * n + s] = acc;
}

// ---------------------------------------------------------------------------
// Termination scalar: t = mean(h_new, axis=0) @ W_term + b_term
// ---------------------------------------------------------------------------
__global__ void term_kernel(const float* __restrict__ hsum, const float* __restrict__ Wt,
                            const float* __restrict__ bt, float* __restrict__ tout, int n) {
    if (threadIdx.x == 0 && blockIdx.x == 0) {
        float inv = 1.0f / (float)n;
        float acc = bt[0];
        for (int k = 0; k < HD; ++k) acc = fmaf(hsum[k] * inv, Wt[k], acc);
        *tout = acc;
    }
}

extern "C" void kernel_launch(void* const* d_in, const int* in_sizes, int n_in,
                              void* d_out, int out_size, void* d_ws, size_t ws_size,
                              hipStream_t stream) {
    const int*   srcs   = (const int*)d_in[0];
    const int*   dsts   = (const int*)d_in[1];
    const float* ew     = (const float*)d_in[2];
    const float* x      = (const float*)d_in[3];
    const float* h      = (const float*)d_in[4];
    const float* W_enc  = (const float*)d_in[5];
    const float* b_enc  = (const float*)d_in[6];
    const float* W_msg  = (const float*)d_in[7];
    const float* b_msg  = (const float*)d_in[8];
    const float* W_upd  = (const float*)d_in[9];
    const float* b_upd  = (const float*)d_in[10];
    const float* W_dec  = (const float*)d_in[11];
    const float* b_dec  = (const float*)d_in[12];
    const float* W_term = (const float*)d_in[13];
    const float* b_term = (const float*)d_in[14];
    const float* W_pred = (const float*)d_in[15];
    const float* b_pred = (const float*)d_in[16];

    const int E_ = in_sizes[0];
    const int N_ = in_sizes[3];   // x is (N,1)

    // Output layout: y[N], p[N*N], h_new[N*HD], t[1]
    float* out   = (float*)d_out;
    float* y     = out;
    float* p     = y + N_;
    float* hnew  = p + (size_t)N_ * N_;
    float* tout  = hnew + (size_t)N_ * HD;

    // Workspace: z[N*HD], mbuf[N*HD], hsum[HD]
    float* z     = (float*)d_ws;
    float* mbuf  = z + (size_t)N_ * HD;
    float* hsum  = mbuf + (size_t)N_ * HD;

    // 1) init scatter-max buffer + mean accumulator
    {
        int nElem = N_ * HD;
        int blocks = (nElem + 255) / 256;
        init_m_kernel<<<blocks, 256, 0, stream>>>(mbuf, hsum, nElem);
    }
    // 2) stream-fill p with -1e9 (dominant 400 MB of traffic)
    fill_p_kernel<<<4096, 256, 0, stream>>>(p, (long long)N_ * N_);
    // 3) encoder
    enc_kernel<<<(N_ + 255) / 256, 256, 0, stream>>>(x, h, W_enc, b_enc, z, N_);
    // 4) message GEMM on WMMA + scatter-max (16 edges per wave, 4 waves per block)
    {
        int tiles  = (E_ + 15) / 16;
        int blocks = (tiles + 3) / 4;
        msg_kernel<<<blocks, 128, 0, stream>>>(srcs, dsts, ew, z, W_msg, b_msg, mbuf, E_);
    }
    // 5) update + decode + mean partials
    upd_kernel<<<(N_ + 255) / 256, 256, 0, stream>>>(z, mbuf, W_upd, b_upd, W_dec, b_dec,
                                                     hnew, y, hsum, N_);
    // 6) predecessor scatter
    pred_kernel<<<(E_ + 255) / 256, 256, 0, stream>>>(srcs, dsts, ew, hnew, W_pred, b_pred,
                                                      p, E_, N_);
    // 7) termination scalar
    term_kernel<<<1, 32, 0, stream>>>(hsum, W_term, b_term, tout, N_);
}